// VanillaRNN_75093208203475
// MI455X (gfx1250) — compile-verified
//
#include <hip/hip_runtime.h>
#include <math.h>
#include <stdint.h>

// VanillaRNN on MI455X (gfx1250):
//   h_{t+1} = tanh(h_t + x_t) @ W^T,  B=32, T=2048, N=512, fp32.
// Grid = 32 blocks == 2 clusters x 16 workgroups.
//   cluster        = batch half (16 rows -> one M=16 WMMA tile)
//   WG-in-cluster  = 32-wide output-feature chunk
// Per step each WG computes a 16x32 tile (K=512) via chained
// V_WMMA_F32_16X16X4_F32. h is exchanged through L2 (d_out doubles as the
// exchange buffer, device-scope), steps fenced by s_wait_storecnt +
// cluster barrier (-3). W preloaded once via TDM tensor_load_to_lds
// (with HW padding for LDS bank-conflict avoidance); x double-buffered by
// dedicated waves via GLOBAL_LOAD_ASYNC_TO_LDS_B128.

#define Bdim 32
#define Tdim 2048
#define Ndim 512
#define TP1  2049
#define PADW 516   // LDS row stride (floats): 516 % 64 == 4 -> 16 lanes hit
                   // 16 disjoint bank pairs on b64 fragment loads.

#if __has_builtin(__builtin_amdgcn_tensor_load_to_lds)
#define HAVE_TDM 1
#else
#define HAVE_TDM 0
#endif

#if __has_builtin(__builtin_amdgcn_global_load_async_to_lds_b128)
#define HAVE_ASYNC 1
#else
#define HAVE_ASYNC 0
#endif

#if HAVE_TDM
#define WMID 2  // TDM pad: +2 dwords after every 256 dwords -> mid-row shift
#else
#define WMID 0
#endif

typedef __attribute__((ext_vector_type(2))) float v2f;
typedef __attribute__((ext_vector_type(4))) float v4f;
typedef __attribute__((ext_vector_type(8))) float v8f;
typedef __attribute__((ext_vector_type(4))) unsigned int v4u;
typedef __attribute__((ext_vector_type(4))) int v4i;
typedef __attribute__((ext_vector_type(8))) int v8i;

// Pointer shapes required by the async-to-LDS builtin:
//   arg0: v4i in addrspace(1) (global), arg1: v4i in addrspace(3) (LDS)
static __device__ __forceinline__ __attribute__((address_space(1))) v4i*
to_g4(const void* p) {
  return (__attribute__((address_space(1))) v4i*)(uintptr_t)p;
}
static __device__ __forceinline__ __attribute__((address_space(3))) v4i*
to_l4(const void* p) {
  // generic LDS pointer -> LDS offset (addr[31:0]) -> AS3 pointer
  return (__attribute__((address_space(3))) v4i*)(uint32_t)(uintptr_t)p;
}

static __device__ __forceinline__ void wait_async0() {
#if __has_builtin(__builtin_amdgcn_s_wait_asynccnt)
  __builtin_amdgcn_s_wait_asynccnt(0);
#else
  asm volatile("s_wait_asynccnt 0x0" ::: "memory");
#endif
}
static __device__ __forceinline__ void wait_tensor0() {
#if __has_builtin(__builtin_amdgcn_s_wait_tensorcnt)
  __builtin_amdgcn_s_wait_tensorcnt(0);
#else
  asm volatile("s_wait_tensorcnt 0x0" ::: "memory");
#endif
}

// Dynamic LDS layout (floats):
//   [0        , 32*PADW)   W chunk: 32 output rows x 512 K (+pad)
//   [32*PADW  , +16*PADW)  A = tanh(h_t + x_t): 16 rows x 512 (+pad)
//   [48*PADW  , +2*16*512) X double buffer
__global__ __launch_bounds__(128)
void rnn_cluster_wmma_kernel(const float* __restrict__ X,
                             const float* __restrict__ h0,
                             const float* __restrict__ W,
                             float* __restrict__ out)
{
  extern __shared__ float smem[];
  float* Wlds = smem;                       // 32*PADW
  float* Alds = smem + 32 * PADW;           // 16*PADW
  float* Xbuf = smem + 48 * PADW;           // 2*16*512

  const int tid  = threadIdx.x;
  const int lane = tid & 31;
  const int wave = tid >> 5;
  const int wg   = blockIdx.x & 15;         // workgroup-in-cluster -> N chunk
  const int grp  = blockIdx.x >> 4;         // cluster -> batch half
  const int r0   = grp * 16;                // batch rows [r0, r0+16)
  const int c0   = wg * 32;                 // output cols [c0, c0+32)

  // ---- one-time: W rows [c0,c0+32) -> LDS ----
#if HAVE_TDM
  if (tid < 32) {  // TDM issue is per-wave; wave 0 drives the DMA
    uint64_t gaddr = (uint64_t)(uintptr_t)(W + (size_t)c0 * Ndim);
    uint32_t laddr = (uint32_t)(uintptr_t)Wlds;
    v4u g0 = {};
    g0.x = 0x1u;                                    // count=1, user D#
    g0.y = laddr;                                   // lds_addr
    g0.z = (uint32_t)gaddr;                         // global_addr[31:0]
    g0.w = ((uint32_t)(gaddr >> 32) & 0x1FFFFFFu)   // global_addr[56:32]
         | (2u << 30);                              // type = 2 ("image")
    v8i g1 = {};
    // data_size=4B (2), pad_enable, pad_interval=256dw (7), pad_amount=2dw (1)
    g1[0] = (int)((2u << 16) | (1u << 20) | (7u << 22) | (1u << 25));
    g1[1] = (int)(512u << 16);  // tensor_dim0[15:0]  = 512 (bits 63:48)
    g1[2] = (int)(512u << 16);  // tensor_dim1[15:0]  = 512 (bits 95:80)
    g1[3] = (int)(512u << 16);  // tile_dim0          = 512 (bits 127:112)
    g1[4] = 32;                 // tile_dim1          = 32  (bits 143:128)
    g1[5] = 512;                // tensor_dim0_stride = 512 (bits 191:160)
    g1[6] = 0;
    g1[7] = 0;
    v4i z4 = {};
    v8i z8 = {};
    __builtin_amdgcn_tensor_load_to_lds(g0, g1, z4, z4, z8, 0);
    wait_tensor0();
  }
#else
  for (int i = tid; i < 32 * 128; i += 128) {
    int row = i >> 7;
    int q   = (i & 127) << 2;
    v4f w = *(const v4f*)(W + (size_t)(c0 + row) * Ndim + q);
    *(v4f*)(Wlds + row * PADW + q) = w;
  }
#endif

  // ---- h0 chunk -> out[:,0,:] (this is also the t=0 exchange data) ----
  for (int i = tid; i < 16 * 32; i += 128) {
    int row  = i >> 5;
    int colg = c0 + (i & 31);
    float v = h0[(size_t)(r0 + row) * Ndim + colg];
    __hip_atomic_store(&out[(size_t)(r0 + row) * TP1 * Ndim + colg], v,
                       __ATOMIC_RELAXED, __HIP_MEMORY_SCOPE_AGENT);
  }
  // ---- x_0 -> Xbuf[0] ----
  for (int i = tid; i < 16 * 128; i += 128) {
    int row = i >> 7;
    int q   = (i & 127) << 2;
    const float* src = X + ((size_t)(r0 + row) * Tdim) * Ndim + q;
#if HAVE_ASYNC
    __builtin_amdgcn_global_load_async_to_lds_b128(
        to_g4(src), to_l4(Xbuf + row * Ndim + q), 0, 0);
#else
    *(v4f*)(Xbuf + row * Ndim + q) = *(const v4f*)src;
#endif
  }
  wait_async0();
  asm volatile("s_wait_storecnt 0x0" ::: "memory");
  __syncthreads();
  __builtin_amdgcn_s_cluster_barrier();

  // Per-lane fragment bases (f32 16x16x4 layout):
  //   A 16x4: lanes 0-15 -> M=lane, VGPR{0,1}=K{0,1}; lanes 16-31 -> K{2,3}
  //   B 4x16: lanes 0-15 -> N=lane, VGPR{0,1}=K{0,1}; lanes 16-31 -> K{2,3}
  const float* Afrag = Alds + (lane & 15) * PADW + ((lane >> 4) << 1);
  const float* Bfrag = Wlds + ((wave & 1) * 16 + (lane & 15)) * PADW
                            + ((lane >> 4) << 1);

  for (int t = 0; t < Tdim; ++t) {
    const float* Xc = Xbuf + (t & 1) * (16 * Ndim);

    // ---- phase A: A = tanh(h_t + x_t); h_t read device-coherently ----
    for (int j = 0; j < 32; ++j) {
      int e   = (tid + (j << 7)) << 1;  // even element index, 2 floats/iter
      int row = e >> 9;
      int col = e & 511;
      union { unsigned long long u; float f[2]; } hv;
      hv.u = __hip_atomic_load(
          (unsigned long long*)&out[(size_t)(r0 + row) * TP1 * Ndim +
                                    (size_t)t * Ndim + col],
          __ATOMIC_RELAXED, __HIP_MEMORY_SCOPE_AGENT);
      v2f xv = *(const v2f*)(Xc + row * Ndim + col);
      v2f r;
      r.x = tanhf(hv.f[0] + xv.x);
      r.y = tanhf(hv.f[1] + xv.y);
      *(v2f*)(Alds + row * PADW + col) = r;
    }
    __syncthreads();

    if (wave < 2) {
      // ---- phase B: 16x16 tile, K=512 -> 128 chained fp32 WMMAs.
      //      Split at K=256: TDM padding shifts W's row tail by WMID dwords.
      v8f c = {};
#pragma unroll 16
      for (int k0 = 0; k0 < 256; k0 += 4) {
        v2f a = *(const v2f*)(Afrag + k0);
        v2f b = *(const v2f*)(Bfrag + k0);
        c = __builtin_amdgcn_wmma_f32_16x16x4_f32(
            false, a, false, b, (short)0, c, false, false);
      }
#pragma unroll 16
      for (int k0 = 256; k0 < 512; k0 += 4) {
        v2f a = *(const v2f*)(Afrag + k0);
        v2f b = *(const v2f*)(Bfrag + k0 + WMID);
        c = __builtin_amdgcn_wmma_f32_16x16x4_f32(
            false, a, false, b, (short)0, c, false, false);
      }
      // ---- phase C: C/D layout -> VGPR v holds M=v (lanes 0-15), M=v+8 ----
      int colg = c0 + wave * 16 + (lane & 15);
      int rb   = r0 + ((lane >> 4) << 3);
      size_t tOff = (size_t)(t + 1) * Ndim;
#pragma unroll
      for (int v = 0; v < 8; ++v) {
        __hip_atomic_store(&out[(size_t)(rb + v) * TP1 * Ndim + tOff + colg],
                           c[v], __ATOMIC_RELAXED, __HIP_MEMORY_SCOPE_AGENT);
      }
    } else {
      // ---- waves 2,3: stream x_{t+1} into the other X buffer ----
      int tn = t + 1;
      if (tn < Tdim) {
        float* Xn = Xbuf + (tn & 1) * (16 * Ndim);
        for (int i = tid - 64; i < 16 * 128; i += 64) {
          int row = i >> 7;
          int q   = (i & 127) << 2;
          const float* src = X + ((size_t)(r0 + row) * Tdim + tn) * Ndim + q;
#if HAVE_ASYNC
          __builtin_amdgcn_global_load_async_to_lds_b128(
              to_g4(src), to_l4(Xn + row * Ndim + q), 0, 0);
#else
          *(v4f*)(Xn + row * Ndim + q) = *(const v4f*)src;
#endif
        }
        if (tn + 1 < Tdim) {  // warm L2 for x_{t+2}
          int l = tid - 64;
          __builtin_prefetch(
              X + ((size_t)(r0 + (l >> 2)) * Tdim + tn + 1) * Ndim
                + ((l & 3) << 7), 0, 1);
        }
      }
    }

    // ---- fence async LDS fills + tile stores, then step barrier ----
    wait_async0();
    asm volatile("s_wait_storecnt 0x0" ::: "memory");
    __syncthreads();
    __builtin_amdgcn_s_cluster_barrier();
  }
}

extern "C" void kernel_launch(void* const* d_in, const int* in_sizes, int n_in,
                              void* d_out, int out_size, void* d_ws,
                              size_t ws_size, hipStream_t stream) {
  const float* X  = (const float*)d_in[0];   // (B, T, N) fp32
  const float* h0 = (const float*)d_in[1];   // (B, N)    fp32
  const float* W  = (const float*)d_in[2];   // (N, N)    fp32
  float* out = (float*)d_out;                // (B, T+1, N) fp32

  const size_t smem =
      (size_t)(48 * PADW + 2 * 16 * Ndim) * sizeof(float);  // ~160.75 KB

  (void)hipFuncSetAttribute(
      reinterpret_cast<const void*>(rnn_cluster_wmma_kernel),
      hipFuncAttributeMaxDynamicSharedMemorySize, (int)smem);

  // 32 blocks == 2 clusters x 16 workgroups (cluster barriers inside).
  rnn_cluster_wmma_kernel<<<dim3(32), dim3(128), smem, stream>>>(X, h0, W, out);
}